// StillingerWeberLayer_67826123538730
// MI455X (gfx1250) — compile-verified
//
#include <hip/hip_runtime.h>
#include <math.h>

typedef __attribute__((ext_vector_type(2))) float v2f;
typedef __attribute__((ext_vector_type(8))) float v8f;

#define SW_EPS 1e-12f

__device__ __forceinline__ float fast_rcp(float x)  { return __builtin_amdgcn_rcpf(x); }
__device__ __forceinline__ float fast_sqrt(float x) { return __builtin_amdgcn_sqrtf(x); }

// One wave32 = one particle; lane j = neighbor j (K must be 32).
// cos-matrix  C' = (diag(1/r) D)(diag(1/r) D)^T - cos_b0   via V_WMMA_F32_16X16X4_F32 (C operand = -cb)
// weight matrix F = f f^T (rank-1 outer product)            via V_WMMA_F32_16X16X4_F32
// E3 pair sum = (full - diagonal)/2 symmetry trick -> branch-free fma loop.
__global__ __launch_bounds__(256) void sw_energy_kernel(
    const int* __restrict__ contrib,
    const float* __restrict__ coords,
    const int* __restrict__ nbrs,
    const float* __restrict__ pA, const float* __restrict__ pB,
    const float* __restrict__ pp, const float* __restrict__ pq,
    const float* __restrict__ psigma, const float* __restrict__ pgamma,
    const float* __restrict__ pcutoff, const float* __restrict__ plam,
    const float* __restrict__ pcb,
    float* __restrict__ partial,
    int n, int k)
{
    __shared__ float wsum[8];
    const int lane = threadIdx.x & 31;
    const int wid  = threadIdx.x >> 5;
    const int i    = (int)((blockIdx.x * blockDim.x + threadIdx.x) >> 5);

    const float A     = *pA;
    const float Bc    = *pB;
    const float p     = *pp;
    const float q     = *pq;
    const float sigma = *psigma;
    const float gam   = *pgamma;
    const float cutf  = *pcutoff;
    const float lam   = *plam;
    const float cb    = *pcb;

    float wave_part = 0.0f;

    if (i < n) {
        // --- gather: own position + neighbor position, per-lane displacement
        const float xi = coords[3 * i + 0];
        const float yi = coords[3 * i + 1];
        const float zi = coords[3 * i + 2];
        const int   nb = nbrs[i * k + lane];
        const float dx = coords[3 * nb + 0] - xi;
        const float dy = coords[3 * nb + 1] - yi;
        const float dz = coords[3 * nb + 2] - zi;

        const float r  = fast_sqrt(dx * dx + dy * dy + dz * dz + SW_EPS);
        const float ir = fast_rcp(r);

        // --- two-body term (matches reference masking semantics)
        const bool  within = r < cutf;
        const float rs     = within ? r : 0.5f * cutf;
        const float sig_r  = sigma * fast_rcp(rs);

        float poly;
        if (p == 4.0f && q == 0.0f) {          // wave-uniform fast path (actual params)
            const float s2 = sig_r * sig_r;
            poly = Bc * s2 * s2 - 1.0f;        // sig_r^0 == 1
        } else {                               // generic fallback
            poly = Bc * powf(sig_r, p) - powf(sig_r, q);
        }
        float e2 = A * poly * __expf(sigma * fast_rcp(rs - cutf));
        e2 = within ? e2 : 0.0f;

        // --- per-neighbor three-body factor: exp(g/(r1-c)+g/(r2-c)) = f_j * f_k
        const float f3 = __expf(gam * fast_rcp(r - cutf));

        // --- unit displacement rows: U = diag(1/r) * D, so U U^T = cos matrix
        const float ux = dx * ir;
        const float uy = dy * ir;
        const float uz = dz * ir;

        // fragments: A 16x4 f32 layout lanes 0-15 -> {K0,K1}, lanes 16-31 -> {K2,K3};
        // B 4x16 mirrors it, so A-frag(rows) == B-frag(cols) for symmetric products.
        const bool hi   = lane >= 16;
        const int  lcol = lane & 15;
        const float uz_lo = __shfl(uz, lcol, 32);
        const float ux_hi = __shfl(ux, lcol + 16, 32);
        const float uy_hi = __shfl(uy, lcol + 16, 32);
        const float f3_hi = __shfl(f3, lcol + 16, 32);

        v2f ufrag0, ufrag1;                 // rows 0..15 / 16..31 of U
        ufrag0.x = hi ? uz_lo : ux;
        ufrag0.y = hi ? 0.0f  : uy;
        ufrag1.x = hi ? uz    : ux_hi;
        ufrag1.y = hi ? 0.0f  : uy_hi;

        v2f ffrag0, ffrag1;                 // rows = [f_j, 0, 0, 0]
        ffrag0.x = hi ? 0.0f : f3;
        ffrag0.y = 0.0f;
        ffrag1.x = hi ? 0.0f : f3_hi;
        ffrag1.y = 0.0f;

        v8f cz  = {0.f, 0.f, 0.f, 0.f, 0.f, 0.f, 0.f, 0.f};
        v8f cnb = {-cb, -cb, -cb, -cb, -cb, -cb, -cb, -cb};

        // cos - cb tiles (C operand pre-loaded with -cb)
        v8f g00 = __builtin_amdgcn_wmma_f32_16x16x4_f32(false, ufrag0, false, ufrag0,
                                                        (short)0, cnb, false, false);
        v8f g01 = __builtin_amdgcn_wmma_f32_16x16x4_f32(false, ufrag0, false, ufrag1,
                                                        (short)0, cnb, false, false);
        v8f g11 = __builtin_amdgcn_wmma_f32_16x16x4_f32(false, ufrag1, false, ufrag1,
                                                        (short)0, cnb, false, false);
        // f_j * f_k outer-product tiles
        v8f F00 = __builtin_amdgcn_wmma_f32_16x16x4_f32(false, ffrag0, false, ffrag0,
                                                        (short)0, cz, false, false);
        v8f F01 = __builtin_amdgcn_wmma_f32_16x16x4_f32(false, ffrag0, false, ffrag1,
                                                        (short)0, cz, false, false);
        v8f F11 = __builtin_amdgcn_wmma_f32_16x16x4_f32(false, ffrag1, false, ffrag1,
                                                        (short)0, cz, false, false);

        // Sum_{j<k} for the two diagonal tiles = (full sum - diagonal)/2; tile01 is all j<k.
        float accS  = 0.0f;   // full sums of tiles (0,0) and (1,1)
        float acc01 = 0.0f;   // tile (0,1)
#pragma unroll
        for (int v = 0; v < 8; ++v) {
            accS  = fmaf(F00[v] * g00[v], g00[v], accS);
            accS  = fmaf(F11[v] * g11[v], g11[v], accS);
            acc01 = fmaf(F01[v] * g01[v], g01[v], acc01);
        }
        // lane-local diagonal correction: j==k term of the symmetric tiles
        const float cd = (ux * ux + uy * uy + uz * uz) - cb;
        const float dg = f3 * f3 * cd * cd;

        float part = 0.5f * e2 + lam * (0.5f * (accS - dg) + acc01);
        // deterministic wave32 reduction
#pragma unroll
        for (int off = 16; off > 0; off >>= 1)
            part += __shfl_xor(part, off, 32);

        const float w = (contrib[i] == 1) ? 1.0f : 0.0f;
        wave_part = part * w;
    }

    if (lane == 0) wsum[wid] = wave_part;
    __syncthreads();
    if (threadIdx.x == 0) {
        float s = 0.0f;
#pragma unroll
        for (int t = 0; t < 8; ++t) s += wsum[t];
        partial[blockIdx.x] = s;
    }
}

__global__ __launch_bounds__(256) void sw_reduce_kernel(const float* __restrict__ partial,
                                                        int m, float* __restrict__ out)
{
    __shared__ float sh[256];
    float s = 0.0f;
    for (int idx = (int)threadIdx.x; idx < m; idx += 256) s += partial[idx];
    sh[threadIdx.x] = s;
    __syncthreads();
    for (int off = 128; off > 0; off >>= 1) {
        if ((int)threadIdx.x < off) sh[threadIdx.x] += sh[threadIdx.x + off];
        __syncthreads();
    }
    if (threadIdx.x == 0) out[0] = sh[0];
}

extern "C" void kernel_launch(void* const* d_in, const int* in_sizes, int n_in,
                              void* d_out, int out_size, void* d_ws, size_t ws_size,
                              hipStream_t stream)
{
    const int*   contrib = (const int*)  d_in[0];
    const float* coords  = (const float*)d_in[1];
    /* d_in[2] = num_neighbors (unused by reference math) */
    const int*   nbrs    = (const int*)  d_in[3];
    const float* pA   = (const float*)d_in[4];
    const float* pB   = (const float*)d_in[5];
    const float* pp   = (const float*)d_in[6];
    const float* pq   = (const float*)d_in[7];
    const float* psig = (const float*)d_in[8];
    const float* pgam = (const float*)d_in[9];
    const float* pcut = (const float*)d_in[10];
    const float* plam = (const float*)d_in[11];
    const float* pcb  = (const float*)d_in[12];

    const int n = in_sizes[0];
    const int k = in_sizes[3] / n;          // 32 per reference
    const int nblocks = (n + 7) / 8;        // 8 waves (particles) per 256-thread block

    float* partial = (float*)d_ws;          // nblocks floats of scratch

    sw_energy_kernel<<<nblocks, 256, 0, stream>>>(
        contrib, coords, nbrs, pA, pB, pp, pq, psig, pgam, pcut, plam, pcb,
        partial, n, k);

    sw_reduce_kernel<<<1, 256, 0, stream>>>(partial, nblocks, (float*)d_out);
}